// Linear_65747359367815
// MI455X (gfx1250) — compile-verified
//
#include <hip/hip_runtime.h>

typedef float v2f __attribute__((ext_vector_type(2)));
typedef float v8f __attribute__((ext_vector_type(8)));
typedef int   v4i __attribute__((vector_size(16)));   // async-copy builtin operand type

#define B_    256
#define N_    4096
#define C_    16
#define W_    10
#define L_    4087            // N - W + 1
#define KN    16              // n per LDS chunk
#define KSEG  128             // n per workgroup
#define NSEG  (N_ / KSEG)     // 32 k-segments
#define NCHUNK (KSEG / KN)    // 8 chunks per segment
#define NBUF  3               // pipeline depth 2 -> triple buffer

#if defined(__has_builtin)
#if __has_builtin(__builtin_amdgcn_global_load_async_to_lds_b128) && \
    __has_builtin(__builtin_amdgcn_s_wait_asynccnt)
#define USE_ASYNC 1
#endif
#endif

// ---------------------------------------------------------------------------
// Stage 1: depthwise valid correlation as GEMM vs band matrix, f32 WMMA,
// triple-buffered LDS with depth-2 async global->LDS prefetch (ASYNCcnt).
// Grid: (B/16, NSEG). Block: 256 threads (8 waves).
// Wave w owns channels {2w, 2w+1} x {real, imag} -> 4 accumulators 16x16 f32.
// ---------------------------------------------------------------------------
__global__ __launch_bounds__(256) void stage1_wmma(
    const float* __restrict__ x,     // [B][N][C][2]
    const float* __restrict__ wr,    // [C][L]
    const float* __restrict__ wi,    // [C][L]
    float* __restrict__ part)        // [NSEG][B][W][C][2]
{
    __shared__ float lx[NBUF][KN][16][36]; // [buf][n][b][cp], padded rows (16B aligned, conflict-free)
    __shared__ float lw[NBUF][2][17][40];  // [buf][p][c][t]; row c==16 is all zeros (B-col padding)

    const int tid  = threadIdx.x;
    const int lane = tid & 31;
    const int wave = tid >> 5;
    const int b0   = blockIdx.x << 4;
    const int kseg = blockIdx.y;
    const int c0   = wave << 1;
    const int m    = lane & 15;          // A row (batch)
    const int kh   = (lane >> 4) << 1;   // K half: lanes 0-15 -> K0/K1, 16-31 -> K2/K3
    const int wcol = lane & 15;          // B col (w)

    // issue the x-tile copy for one chunk into buffer `buf`
    auto issue_x = [&](int nbase, int buf) {
        #pragma unroll
        for (int i = 0; i < 8; ++i) {
            int lin = i * 256 + tid;             // 2048 float4s
            int bl  = lin >> 7;
            int rem = lin & 127;
            int nl  = rem >> 3;
            int q   = rem & 7;
            const float* g = x + (((size_t)(b0 + bl) * N_ + (size_t)(nbase + nl)) << 5) + (q << 2);
            float* l = &lx[buf][nl][bl][q << 2];
#ifdef USE_ASYNC
            __builtin_amdgcn_global_load_async_to_lds_b128((v4i*)g, (v4i*)l, 0, 0);
#else
            *(float4*)l = *(const float4*)g;
#endif
        }
    };
    // stage the zero-padded filter window for one chunk
    auto fill_w = [&](int nbase, int buf) {
        #pragma unroll
        for (int i = 0; i < 5; ++i) {
            int lin = i * 256 + tid;             // 1280 = 2*16*40
            int p = lin / 640;
            int r = lin - p * 640;
            int c = r / 40;
            int t = r - c * 40;
            int gi = nbase - 17 + t;
            const float* wf = p ? wi : wr;
            lw[buf][p][c][t] = (gi >= 0 && gi < L_) ? wf[c * L_ + gi] : 0.0f;
        }
    };

    // zero rows for B-column padding (written once, never overwritten)
    if (tid < 80 * NBUF) {
        int buf = tid / 80;
        int p   = (tid / 40) & 1;
        int t   = tid % 40;
        lw[buf][p][16][t] = 0.0f;
    }

    // lanes with wcol>=10 read the zero row: no selects in the hot loop
    const int rowA  = (wcol < W_) ? c0 : 16;
    const int rowB  = (wcol < W_) ? (c0 + 1) : 16;
    const int ubase = 17 - wcol;                 // + ka -> filter index (always in [2,33])

    v8f acc[4] = {};                             // (c0,re),(c0,im),(c0+1,re),(c0+1,im)

    // prologue: prefetch chunks 0 and 1
    issue_x(kseg * KSEG + 0 * KN, 0);
    fill_w(kseg * KSEG + 0 * KN, 0);
    issue_x(kseg * KSEG + 1 * KN, 1);
    fill_w(kseg * KSEG + 1 * KN, 1);

    for (int chunk = 0; chunk < NCHUNK; ++chunk) {
        const int cur = chunk % NBUF;
        if (chunk + 2 < NCHUNK) {
            const int pb = (chunk + 2) % NBUF;   // buffer freed at previous tail barrier
            issue_x(kseg * KSEG + (chunk + 2) * KN, pb);
            fill_w(kseg * KSEG + (chunk + 2) * KN, pb);
#ifdef USE_ASYNC
            __builtin_amdgcn_s_wait_asynccnt(16);  // 3 chunks out -> wait for current (in-order)
#endif
        } else if (chunk + 1 < NCHUNK) {
#ifdef USE_ASYNC
            __builtin_amdgcn_s_wait_asynccnt(8);   // 2 chunks out
#endif
        } else {
#ifdef USE_ASYNC
            __builtin_amdgcn_s_wait_asynccnt(0);   // last chunk
#endif
        }
        __syncthreads();                           // all waves' data visible

        for (int k0 = 0; k0 < KN; k0 += 4) {
            const int ka = k0 + kh;
            #pragma unroll
            for (int j = 0; j < 4; ++j) {
                const int c   = c0 + (j >> 1);
                const int p   = j & 1;
                const int cp  = (c << 1) + p;
                const int row = (j >> 1) ? rowB : rowA;
                v2f a, b;
                a.x = lx[cur][ka][m][cp];
                a.y = lx[cur][ka + 1][m][cp];
                b.x = lw[cur][p][row][ubase + ka];
                b.y = lw[cur][p][row][ubase + ka + 1];
                acc[j] = __builtin_amdgcn_wmma_f32_16x16x4_f32(
                    false, a, false, b, (short)0, acc[j], false, false);
            }
        }
        __syncthreads();                           // done reading `cur` before it is refilled
    }

    // store partials: C/D layout lane(0-15): M=r, N=lane; lane(16-31): M=8+r
    if (wcol < W_) {
        #pragma unroll
        for (int j = 0; j < 4; ++j) {
            const int c = c0 + (j >> 1);
            const int p = j & 1;
            #pragma unroll
            for (int r = 0; r < 8; ++r) {
                const int mm = r + ((lane >> 4) << 3);
                const size_t idx =
                    (((size_t)(kseg * B_ + b0 + mm) * W_ + wcol) << 5) + (c << 1) + p;
                part[idx] = acc[j][r];
            }
        }
    }
}

// ---------------------------------------------------------------------------
// Stage 2+3: reduce k-segments, |f|^2 * f, Linear(2C->2), output filters.
// Grid: B blocks of 32 threads.
// ---------------------------------------------------------------------------
__global__ __launch_bounds__(32) void stage23(
    const float* __restrict__ part,   // [NSEG][B][W][C][2]
    const float* __restrict__ Wnl,    // [C][2][2C]
    const float* __restrict__ Wor,    // [C][W]
    const float* __restrict__ Woi,    // [C][W]
    float* __restrict__ outpre)       // [B][C][2]
{
    __shared__ float tf[W_][32];      // tf[w][i], i = p*16 + c  (concat [fr, fi])
    const int b = blockIdx.x;
    const int t = threadIdx.x;

    #pragma unroll
    for (int w = 0; w < W_; ++w) {    // thread t owns cp = t
        float s = 0.0f;
        for (int ks = 0; ks < NSEG; ++ks)
            s += part[(((size_t)(ks * B_ + b) * W_ + w) << 5) + t];
        const int c = t >> 1, p = t & 1;
        tf[w][p * 16 + c] = s;
    }
    __syncthreads();
    // amplitude scaling: each thread handles 5 (w,c) pairs (owns both re & im)
    #pragma unroll
    for (int j = 0; j < 5; ++j) {
        const int idx = j * 32 + t;   // 0..159
        const int w = idx >> 4;
        const int c = idx & 15;
        float fr = tf[w][c], fi = tf[w][16 + c];
        float amp = fr * fr + fi * fi;
        tf[w][c]      = amp * fr;
        tf[w][16 + c] = amp * fi;
    }
    __syncthreads();

    const int c = t >> 1, o = t & 1;
    const float* wo = o ? Woi : Wor;
    float acc = 0.0f;
    #pragma unroll
    for (int w = 0; w < W_; ++w) {
        float s = 0.0f;
        #pragma unroll
        for (int i = 0; i < 32; ++i)
            s += tf[w][i] * Wnl[c * 64 + o * 32 + i];
        acc += s * wo[c * W_ + w];
    }
    outpre[(b << 5) + (c << 1) + o] = acc;
}

// ---------------------------------------------------------------------------
// BatchNorm1d (training stats over (B, 2) per channel). Grid: C blocks x 256.
// ---------------------------------------------------------------------------
__global__ __launch_bounds__(256) void bn_kernel(
    const float* __restrict__ outpre,
    const float* __restrict__ gamma,
    const float* __restrict__ beta,
    float* __restrict__ out)          // [B][C][2]
{
    __shared__ float ssum[256];
    __shared__ float ssq[256];
    const int c = blockIdx.x, t = threadIdx.x;
    const float v0 = outpre[(t << 5) + (c << 1) + 0];
    const float v1 = outpre[(t << 5) + (c << 1) + 1];
    ssum[t] = v0 + v1;
    ssq[t]  = v0 * v0 + v1 * v1;
    __syncthreads();
    for (int s = 128; s > 0; s >>= 1) {
        if (t < s) { ssum[t] += ssum[t + s]; ssq[t] += ssq[t + s]; }
        __syncthreads();
    }
    const float mean = ssum[0] * (1.0f / 512.0f);
    const float var  = ssq[0] * (1.0f / 512.0f) - mean * mean;
    const float inv  = rsqrtf(var + 1e-5f) * gamma[c];
    out[(t << 5) + (c << 1) + 0] = (v0 - mean) * inv + beta[c];
    out[(t << 5) + (c << 1) + 1] = (v1 - mean) * inv + beta[c];
}

extern "C" void kernel_launch(void* const* d_in, const int* in_sizes, int n_in,
                              void* d_out, int out_size, void* d_ws, size_t ws_size,
                              hipStream_t stream) {
    const float* x   = (const float*)d_in[0];
    const float* wr  = (const float*)d_in[1];
    const float* wi  = (const float*)d_in[2];
    const float* wnl = (const float*)d_in[3];
    const float* wor = (const float*)d_in[4];
    const float* woi = (const float*)d_in[5];
    const float* gam = (const float*)d_in[6];
    const float* bet = (const float*)d_in[7];
    float* out = (float*)d_out;

    float* part   = (float*)d_ws;                                   // NSEG*B*W*C*2 = 2,621,440 f32
    float* outpre = part + (size_t)NSEG * B_ * W_ * C_ * 2;         // B*C*2 = 8192 f32

    stage1_wmma<<<dim3(B_ / 16, NSEG), 256, 0, stream>>>(x, wr, wi, part);
    stage23<<<B_, 32, 0, stream>>>(part, wnl, wor, woi, outpre);
    bn_kernel<<<C_, 256, 0, stream>>>(outpre, gam, bet, out);
}